// Signature_48361331753749
// MI455X (gfx1250) — compile-verified
//
#include <hip/hip_runtime.h>

// Depth-4 streamed path signature (S=8192, d=8) via 3-level parallel Chen scan.
// Bandwidth bound: 153 MB output / 23.3 TB/s ~= 6.6 us floor; compute ~0.44 GFLOP.
// Level-4 update per step expressed as 64x64 <- (64x20)x(20x64) matmul and driven
// by V_WMMA_F32_16X16X4_F32 (5 K-chunks x 16 tiles, 2 tiles per wave).
// Fragments built branchlessly from loop-invariant 0/1 lane masks (no EXEC games).

#define S_LEN   8192
#define NB      128     // blocks
#define LB      64      // steps per block (S_LEN / NB)
#define NSUP    16      // super-blocks
#define SUPLEN  8       // blocks per super-block (NB / NSUP)
#define SIG_SZ  4680    // 8 + 64 + 512 + 4096
#define L2_OFF  8
#define L3_OFF  72
#define L4_OFF  584

typedef float v2f __attribute__((ext_vector_type(2)));
typedef float v8f __attribute__((ext_vector_type(8)));

// ---------------------------------------------------------------------------
// Generic Chen product element: C = A (.) B, flat index e over [lvl1..lvl4]
// ---------------------------------------------------------------------------
__device__ __forceinline__ float chen_elem(int e, const float* A, const float* B) {
  if (e < L2_OFF) {                                   // level 1
    return A[e] + B[e];
  } else if (e < L3_OFF) {                            // level 2
    int q = e - L2_OFF; int i = q >> 3, j = q & 7;
    return A[e] + B[e] + A[i] * B[j];
  } else if (e < L4_OFF) {                            // level 3
    int q = e - L3_OFF; int i = q >> 6, j = (q >> 3) & 7, k = q & 7;
    return A[e] + B[e]
         + A[i] * B[L2_OFF + ((j << 3) | k)]
         + A[L2_OFF + ((i << 3) | j)] * B[k];
  } else {                                            // level 4
    int q = e - L4_OFF; int i = q >> 9, j = (q >> 6) & 7, k = (q >> 3) & 7, l = q & 7;
    return A[e] + B[e]
         + A[i] * B[L3_OFF + ((j << 6) | (k << 3) | l)]
         + A[L2_OFF + ((i << 3) | j)] * B[L2_OFF + ((k << 3) | l)]
         + A[L3_OFF + ((i << 6) | (j << 3) | k)] * B[l];
  }
}

// ---------------------------------------------------------------------------
// Per-step increment powers into LDS: z, zz = z(x)z, z3 = z(x)z(x)z / 6
// ---------------------------------------------------------------------------
__device__ __forceinline__ void fill_z(const float* __restrict__ x, int pos, int tid,
                                       float* sZ, float* sZZ, float* sZ3) {
  if (tid < 8) {
    float cur  = x[pos * 8 + tid];
    float prev = (pos > 0) ? x[(pos - 1) * 8 + tid] : 0.0f;
    sZ[tid] = cur - prev;
  }
  __syncthreads();
  if (tid < 64) sZZ[tid] = sZ[tid >> 3] * sZ[tid & 7];
#pragma unroll
  for (int t = 0; t < 2; ++t) {
    int e = tid + t * 256;
    int i = e >> 6, j = (e >> 3) & 7, k = e & 7;
    sZ3[e] = sZ[i] * sZ[j] * sZ[k] * (1.0f / 6.0f);
  }
  __syncthreads();
}

// ---------------------------------------------------------------------------
// Loop-invariant per-lane tile context: row/col indices and 0/1 delta masks.
//   A frag (16x4): lanes 0-15 M=row, VGPR0=K0/VGPR1=K1; lanes 16-31 K2/K3.
//   B frag (4x16): lanes hold N=lane%16; VGPR0=K0|K2, VGPR1=K1|K3.
// ---------------------------------------------------------------------------
struct TileInv {
  int r, c;                          // global row (i,j) / col (k,l)
  float mE0, mE1, mE2, mE3;          // (ck == q) for q = 2h,2h+1,4+2h,5+2h
  float mF0, mF1, mF2, mF3;          // (rj == q)
  float hm;                          // half==0 (K=16..19 tail lives in lanes 0-15)
};

__device__ __forceinline__ TileInv make_tile_inv(int tile, int lane) {
  TileInv T;
  const int tr = tile >> 2, tc = tile & 3, half = lane >> 4;
  T.r = tr * 16 + (lane & 15);
  T.c = tc * 16 + (lane & 15);
  const int rj = T.r & 7, ck = T.c >> 3;
  const int q0 = 2 * half, q1 = q0 + 1, q2 = q0 + 4, q3 = q2 + 1;
  T.mE0 = (ck == q0) ? 1.0f : 0.0f;  T.mE1 = (ck == q1) ? 1.0f : 0.0f;
  T.mE2 = (ck == q2) ? 1.0f : 0.0f;  T.mE3 = (ck == q3) ? 1.0f : 0.0f;
  T.mF0 = (rj == q0) ? 1.0f : 0.0f;  T.mF1 = (rj == q1) ? 1.0f : 0.0f;
  T.mF2 = (rj == q2) ? 1.0f : 0.0f;  T.mF3 = (rj == q3) ? 1.0f : 0.0f;
  T.hm  = (half == 0) ? 1.0f : 0.0f;
  return T;
}

// Level-4 update: M[(ij),(kl)] += [Q|F|G2|zz/24] . [E;R;z2^T;zz^T] (K=18, pad 20)
__device__ __forceinline__ v8f wmma_step(v8f acc, const TileInv& T, int half,
    const float* sG1, const float* sG2, const float* sG3,
    const float* sZ, const float* sZZ, const float* sZ3) {
  const int ri = T.r >> 3, cl = T.c & 7;
  const float aG1  = sG1[ri];
  const float aG2  = sG2[T.r];
  const float aZZ  = sZZ[T.r] * (1.0f / 24.0f);
  const float bZ   = sZ[cl];
  const float bZZ  = sZZ[T.c];
  const float2* g3p = (const float2*)(sG3 + T.r * 8);
  const float2 g01 = g3p[half];          // Q cols q = 2h, 2h+1
  const float2 g23 = g3p[2 + half];      // Q cols q = 4+2h, 5+2h
  const int zb = 2 * half;
  const float z30 = sZ3[(zb + 0) * 64 + T.c];   // R rows q-8 = 2h, 2h+1
  const float z31 = sZ3[(zb + 1) * 64 + T.c];
  const float z32 = sZ3[(zb + 4) * 64 + T.c];   // R rows q-8 = 4+2h, 5+2h
  const float z33 = sZ3[(zb + 5) * 64 + T.c];

  v2f a, b;
  // K 0..3 : G3 (x) z
  a.x = g01.x;            a.y = g01.y;
  b.x = bZ * T.mE0;       b.y = bZ * T.mE1;
  acc = __builtin_amdgcn_wmma_f32_16x16x4_f32(false, a, false, b, (short)0, acc, false, false);
  // K 4..7
  a.x = g23.x;            a.y = g23.y;
  b.x = bZ * T.mE2;       b.y = bZ * T.mE3;
  acc = __builtin_amdgcn_wmma_f32_16x16x4_f32(false, a, false, b, (short)0, acc, false, false);
  // K 8..11 : G1 (x) z^3/6
  a.x = aG1 * T.mF0;      a.y = aG1 * T.mF1;
  b.x = z30;              b.y = z31;
  acc = __builtin_amdgcn_wmma_f32_16x16x4_f32(false, a, false, b, (short)0, acc, false, false);
  // K 12..15
  a.x = aG1 * T.mF2;      a.y = aG1 * T.mF3;
  b.x = z32;              b.y = z33;
  acc = __builtin_amdgcn_wmma_f32_16x16x4_f32(false, a, false, b, (short)0, acc, false, false);
  // K 16..19 : G2 (x) z^2/2 + z^4/24  (A side masked; upper-half lanes give 0)
  a.x = aG2 * T.hm;       a.y = aZZ * T.hm;
  b.x = bZZ * 0.5f;       b.y = bZZ;
  acc = __builtin_amdgcn_wmma_f32_16x16x4_f32(false, a, false, b, (short)0, acc, false, false);
  return acc;
}

// Lower-level (1..3) state update: G3 += G2(x)z + G1(x)z^2/2 + z^3/6, etc.
__device__ __forceinline__ void lower_update(int tid, float* sG1, float* sG2, float* sG3,
    const float* sZ, const float* sZZ, const float* sZ3) {
  float u3[2];
#pragma unroll
  for (int t = 0; t < 2; ++t) {
    int e = tid + t * 256;
    int i = e >> 6, j = (e >> 3) & 7, k = e & 7, ij = e >> 3;
    u3[t] = sG3[e] + sG2[ij] * sZ[k] + sG1[i] * sZZ[(j << 3) | k] * 0.5f + sZ3[e];
  }
  float u2 = 0.0f, u1 = 0.0f;
  if (tid < 64) u2 = sG2[tid] + sG1[tid >> 3] * sZ[tid & 7] + sZZ[tid] * 0.5f;
  if (tid < 8)  u1 = sG1[tid] + sZ[tid];
  __syncthreads();                 // all reads of old state done
  sG3[tid] = u3[0]; sG3[tid + 256] = u3[1];
  if (tid < 64) sG2[tid] = u2;
  if (tid < 8)  sG1[tid] = u1;
  __syncthreads();                 // new state visible
}

// C/D accumulator layout: VGPR v -> M = v + 8*(lane/16), N = lane%16.
__device__ __forceinline__ void store_acc(float* dst, v8f acc, int tile, int lane) {
  const int tr = tile >> 2, tc = tile & 3, half = lane >> 4, n = lane & 15;
#pragma unroll
  for (int v = 0; v < 8; ++v) {
    int m = v + 8 * half;
    dst[(tr * 16 + m) * 64 + tc * 16 + n] = acc[v];
  }
}

__device__ __forceinline__ v8f init_acc(int tile, int lane, const float* sA, const float* sB) {
  v8f acc;
  const int tr = tile >> 2, tc = tile & 3, half = lane >> 4, n = lane & 15;
#pragma unroll
  for (int v = 0; v < 8; ++v) {
    int m = v + 8 * half;
    int R = tr * 16 + m, C = tc * 16 + n;
    acc[v] = chen_elem(L4_OFF + R * 64 + C, sA, sB);
  }
  return acc;
}

// ---------------------------------------------------------------------------
// K1: per-block segment signature (identity start), result -> ws sigs[blk]
// ---------------------------------------------------------------------------
__global__ void __launch_bounds__(256)
k_block_sig(const float* __restrict__ x, float* __restrict__ sigs) {
  __shared__ __align__(16) float sG3[512];
  __shared__ float sG1[8], sG2[64];
  __shared__ float sZ[8], sZZ[64], sZ3[512];
  const int tid = threadIdx.x, lane = tid & 31, wave = tid >> 5, half = lane >> 4;
  const int t0 = 2 * wave, t1 = t0 + 1;
  const int blk = blockIdx.x;
  const TileInv T0 = make_tile_inv(t0, lane);
  const TileInv T1 = make_tile_inv(t1, lane);

  if (tid < 8)  sG1[tid] = 0.0f;
  if (tid < 64) sG2[tid] = 0.0f;
  sG3[tid] = 0.0f; sG3[tid + 256] = 0.0f;
  v8f acc0 = {}; v8f acc1 = {};
  __syncthreads();

  for (int p = 0; p < LB; ++p) {
    int pos = blk * LB + p;
    if (pos + 1 < S_LEN) __builtin_prefetch(&x[(pos + 1) * 8], 0, 3);
    fill_z(x, pos, tid, sZ, sZZ, sZ3);
    acc0 = wmma_step(acc0, T0, half, sG1, sG2, sG3, sZ, sZZ, sZ3);
    acc1 = wmma_step(acc1, T1, half, sG1, sG2, sG3, sZ, sZZ, sZ3);
    lower_update(tid, sG1, sG2, sG3, sZ, sZZ, sZ3);
  }

  float* g = sigs + (size_t)blk * SIG_SZ;
  if (tid < 8)  g[tid] = sG1[tid];
  if (tid < 64) g[L2_OFF + tid] = sG2[tid];
  g[L3_OFF + tid] = sG3[tid]; g[L3_OFF + tid + 256] = sG3[tid + 256];
  store_acc(g + L4_OFF, acc0, t0, lane);
  store_acc(g + L4_OFF, acc1, t1, lane);
}

// ---------------------------------------------------------------------------
// K2/K3: sequential exclusive Chen scan over `nitems` signatures, in place.
// ---------------------------------------------------------------------------
__global__ void __launch_bounds__(256)
k_scan(float* __restrict__ base, float* __restrict__ total, int nitems) {
  __shared__ float sA[SIG_SZ];
  __shared__ float sB[SIG_SZ];
  const int tid = threadIdx.x;
  float* seg = base + (size_t)blockIdx.x * nitems * SIG_SZ;
  for (int e = tid; e < SIG_SZ; e += 256) sA[e] = 0.0f;   // identity
  __syncthreads();
  for (int it = 0; it < nitems; ++it) {
    float* g = seg + (size_t)it * SIG_SZ;
    for (int e = tid; e < SIG_SZ; e += 256) {
      float bv = g[e];
      g[e] = sA[e];          // write exclusive prefix back in place
      sB[e] = bv;
    }
    __syncthreads();
    float cbuf[19];
    int idx = 0;
    for (int e = tid; e < SIG_SZ; e += 256) cbuf[idx++] = chen_elem(e, sA, sB);
    __syncthreads();
    idx = 0;
    for (int e = tid; e < SIG_SZ; e += 256) sA[e] = cbuf[idx++];
    __syncthreads();
  }
  if (total) {
    for (int e = tid; e < SIG_SZ; e += 256) total[e] = sA[e];
  }
}

// ---------------------------------------------------------------------------
// K4: block b forms P = superExcl (.) localExcl, replays its 64 steps, and
// emits every prefix row.  Level-4 rows staged through LDS for b128 stores.
// ---------------------------------------------------------------------------
__global__ void __launch_bounds__(256)
k_emit(const float* __restrict__ x, const float* __restrict__ sigs,
       const float* __restrict__ supers, float* __restrict__ out) {
  __shared__ __align__(16) float sA[SIG_SZ];   // init buffer, then lvl-4 staging
  __shared__ float sB[SIG_SZ];
  __shared__ __align__(16) float sG3[512];
  __shared__ float sG1[8], sG2[64];
  __shared__ float sZ[8], sZZ[64], sZ3[512];
  const int tid = threadIdx.x, lane = tid & 31, wave = tid >> 5, half = lane >> 4;
  const int t0 = 2 * wave, t1 = t0 + 1;
  const int blk = blockIdx.x;
  const TileInv T0 = make_tile_inv(t0, lane);
  const TileInv T1 = make_tile_inv(t1, lane);

  const float* Ag = supers + (size_t)(blk / SUPLEN) * SIG_SZ;  // exclusive super prefix
  const float* Bg = sigs   + (size_t)blk * SIG_SZ;             // exclusive local prefix
  for (int e = tid; e < SIG_SZ; e += 256) { sA[e] = Ag[e]; sB[e] = Bg[e]; }
  __syncthreads();

  if (tid < 8) sG1[tid] = sA[tid] + sB[tid];
  if (tid < 64) {
    int i = tid >> 3, j = tid & 7;
    sG2[tid] = sA[L2_OFF + tid] + sB[L2_OFF + tid] + sA[i] * sB[j];
  }
#pragma unroll
  for (int t = 0; t < 2; ++t) {
    int e = tid + t * 256;
    sG3[e] = chen_elem(L3_OFF + e, sA, sB);
  }
  v8f acc0 = init_acc(t0, lane, sA, sB);
  v8f acc1 = init_acc(t1, lane, sA, sB);
  __syncthreads();                      // sA free for reuse as staging after this

  float* out1 = out;
  float* out2 = out  + (size_t)S_LEN * 8;
  float* out3 = out2 + (size_t)S_LEN * 64;
  float* out4 = out3 + (size_t)S_LEN * 512;
  float* sStage = sA;                   // 4096-float lvl-4 staging buffer

  for (int p = 0; p < LB; ++p) {
    int pos = blk * LB + p;
    if (pos + 1 < S_LEN) __builtin_prefetch(&x[(pos + 1) * 8], 0, 3);
    fill_z(x, pos, tid, sZ, sZZ, sZ3);
    acc0 = wmma_step(acc0, T0, half, sG1, sG2, sG3, sZ, sZZ, sZ3);
    acc1 = wmma_step(acc1, T1, half, sG1, sG2, sG3, sZ, sZZ, sZ3);
    lower_update(tid, sG1, sG2, sG3, sZ, sZZ, sZ3);

    // ---- emit streamed prefix row `pos` ----
    if (tid < 8)  out1[(size_t)pos * 8  + tid] = sG1[tid];
    if (tid < 64) out2[(size_t)pos * 64 + tid] = sG2[tid];
    out3[(size_t)pos * 512 + tid]       = sG3[tid];
    out3[(size_t)pos * 512 + tid + 256] = sG3[tid + 256];
    // level 4: accumulator -> LDS -> full-cacheline b128 stores
    store_acc(sStage, acc0, t0, lane);
    store_acc(sStage, acc1, t1, lane);
    __syncthreads();
    const float4* st4 = (const float4*)sStage;
    float4* o4 = (float4*)(out4 + (size_t)pos * 4096);
#pragma unroll
    for (int s = 0; s < 4; ++s) o4[tid + 256 * s] = st4[tid + 256 * s];
  }
}

// ---------------------------------------------------------------------------
extern "C" void kernel_launch(void* const* d_in, const int* in_sizes, int n_in,
                              void* d_out, int out_size, void* d_ws, size_t ws_size,
                              hipStream_t stream) {
  (void)in_sizes; (void)n_in; (void)out_size; (void)ws_size;
  const float* x = (const float*)d_in[0];
  float* out     = (float*)d_out;
  float* ws_sigs  = (float*)d_ws;                       // NB  * SIG_SZ floats
  float* ws_super = ws_sigs + (size_t)NB * SIG_SZ;      // NSUP * SIG_SZ floats

  k_block_sig<<<NB, 256, 0, stream>>>(x, ws_sigs);
  k_scan<<<NSUP, 256, 0, stream>>>(ws_sigs, ws_super, SUPLEN);  // local excl + super sigs
  k_scan<<<1,    256, 0, stream>>>(ws_super, nullptr, NSUP);    // super excl
  k_emit<<<NB, 256, 0, stream>>>(x, ws_sigs, ws_super, out);
}